// CfCModel_60696477827202
// MI455X (gfx1250) — compile-verified
//
#include <hip/hip_runtime.h>
#include <hip/hip_bf16.h>

// CfC recurrence, reduced to the only batch row that matters (b=255).
// Phase 1: Pre[T,128] = X[255] @ W_bb[:64,:] + b_bb   (WMMA f32 16x16x4 GEMM)
// Phase 2: single-workgroup latency-optimized scan over T=2048 steps,
//          weights resident in LDS, 3 barriers/step, readout at the end.

#define T_STEPS 2048
#define IN_DIM  64
#define UNITS   50
#define BB_DIM  128
#define HEADS   200          // ff1|ff2|ta|tb concatenated
#define SCAN_THREADS 1024

typedef __attribute__((ext_vector_type(2))) float v2f;
typedef __attribute__((ext_vector_type(8))) float v8f;

// ---------------------------------------------------------------------------
// Phase 1: Pre = X255 @ Wx + b_bb via V_WMMA_F32_16X16X4_F32.
// Tiles: M = 2048/16 = 128, N = 128/16 = 8  -> 1024 waves -> 128 blocks x 8 waves.
// A (16x4 f32): lanes 0-15 = rows M, VGPR0/1 = K,K+1; lanes 16-31 = K+2,K+3.
// B (4x16 f32): mirrored, lane = N column.
// C/D (16x16 f32): VGPR r -> (M = r | r+8, N = lane&15). Bias preloaded into C.
// ---------------------------------------------------------------------------
__global__ void cfc_pre_wmma(const float* __restrict__ x255,
                             const float* __restrict__ Wbb,   // [114,128], rows 0..63 used
                             const float* __restrict__ b_bb,  // [128]
                             float* __restrict__ Pre)         // [2048,128]
{
    const int wave  = (blockIdx.x * blockDim.x + threadIdx.x) >> 5;
    const int lane  = threadIdx.x & 31;
    const int tM    = wave >> 3;          // 0..127
    const int tN    = wave & 7;           // 0..7
    const int m0    = tM << 4;
    const int n0    = tN << 4;
    const int lh    = lane & 15;
    const bool hi   = (lane >= 16);

    // Accumulator initialized with the bias (broadcast down the M dimension).
    v8f c;
    {
        const float bv = b_bb[n0 + lh];
        #pragma unroll
        for (int r = 0; r < 8; ++r) c[r] = bv;
    }

    const float* arow = x255 + (m0 + lh) * IN_DIM;   // X[255][m0+lh][*]

    #pragma unroll
    for (int k = 0; k < IN_DIM; k += 4) {
        const int ka = k + (hi ? 2 : 0);
        v2f a;
        a.x = arow[ka + 0];
        a.y = arow[ka + 1];
        v2f b;
        b.x = Wbb[(ka + 0) * BB_DIM + n0 + lh];
        b.y = Wbb[(ka + 1) * BB_DIM + n0 + lh];
        // 8 args: (neg_a, A, neg_b, B, c_mod, C, reuse_a, reuse_b)
        c = __builtin_amdgcn_wmma_f32_16x16x4_f32(
                false, a, false, b, (short)0, c, false, false);
    }

    #pragma unroll
    for (int r = 0; r < 8; ++r) {
        const int row = m0 + r + (hi ? 8 : 0);
        const int col = n0 + lh;
        Pre[row * BB_DIM + col] = c[r];
    }
}

// ---------------------------------------------------------------------------
// Saturation-safe transcendentals on the v_exp_f32 path.
// ---------------------------------------------------------------------------
__device__ __forceinline__ float fast_tanh(float x) {
    const float e = __expf(2.0f * x);     // v_exp_f32
    return 1.0f - 2.0f / (e + 1.0f);      // exact limits at +/-inf, no NaN
}
__device__ __forceinline__ float fast_sigmoid(float x) {
    return 1.0f / (1.0f + __expf(-x));
}

// ---------------------------------------------------------------------------
// Phase 2: sequential scan, one workgroup (1024 threads = 32 waves).
// LDS: Whh[50][128] + Wcat[128][200] + bias[200] + h[64] + bb[128] + part[4][200]
//    = 33192 floats = 129.7 KB dynamic LDS (WGP has 320 KB).
// Per step: S2 (128 thr, 50-FMA chain) -> B -> S4 (800 thr, 32-FMA split-K
// chains) -> B -> S5 (50 thr, reduce + gates) -> B.   3 barriers/step.
// ---------------------------------------------------------------------------
__global__ void cfc_scan(const float* __restrict__ Pre,
                         const float* __restrict__ Wbb,    // [114,128]
                         const float* __restrict__ Wff1, const float* __restrict__ bff1,
                         const float* __restrict__ Wff2, const float* __restrict__ bff2,
                         const float* __restrict__ Wta,  const float* __restrict__ bta,
                         const float* __restrict__ Wtb,  const float* __restrict__ btb,
                         const float* __restrict__ Wout, const float* __restrict__ bout,
                         float* __restrict__ out)
{
    extern __shared__ float lds[];
    float* Whh   = lds;                       // 50*128
    float* Wcat  = Whh  + UNITS * BB_DIM;     // 128*200
    float* biasC = Wcat + BB_DIM * HEADS;     // 200
    float* hS    = biasC + HEADS;             // 64 (50 used)
    float* bbS   = hS + 64;                   // 128
    float* part  = bbS + BB_DIM;              // 4*200

    const int tid = threadIdx.x;

    // ---- Stage 0: stage weights into LDS ----
    for (int i = tid; i < UNITS * BB_DIM; i += SCAN_THREADS)
        Whh[i] = Wbb[IN_DIM * BB_DIM + i];    // rows 64..113 of W_bb

    for (int i = tid; i < BB_DIM * HEADS; i += SCAN_THREADS) {
        const int k = i / HEADS, o = i % HEADS;
        float v;
        if      (o < 50)  v = Wff1[k * UNITS + o];
        else if (o < 100) v = Wff2[k * UNITS + (o - 50)];
        else if (o < 150) v = Wta [k * UNITS + (o - 100)];
        else              v = Wtb [k * UNITS + (o - 150)];
        Wcat[i] = v;
    }
    if (tid < HEADS) {
        float v;
        if      (tid < 50)  v = bff1[tid];
        else if (tid < 100) v = bff2[tid - 50];
        else if (tid < 150) v = bta [tid - 100];
        else                v = btb [tid - 150];
        biasC[tid] = v;
    }
    if (tid < 64) hS[tid] = 0.0f;   // h0 = 0
    __syncthreads();

    float preReg = (tid < BB_DIM) ? Pre[tid] : 0.0f;

    for (int t = 0; t < T_STEPS; ++t) {
        // Issue next Pre row load + prefetch before the serial chain.
        float preNext = 0.0f;
        if (tid < BB_DIM && (t + 1) < T_STEPS) {
            preNext = Pre[(t + 1) * BB_DIM + tid];
            if ((t + 2) < T_STEPS)
                __builtin_prefetch(Pre + (t + 2) * BB_DIM + tid, 0, 1); // global_prefetch_b8
        }

        // ---- S2: bb = lecun_tanh(Pre[t] + h @ Whh) ----
        if (tid < BB_DIM) {
            float acc = preReg;
            #pragma unroll
            for (int i = 0; i < UNITS; ++i)
                acc = fmaf(hS[i], Whh[i * BB_DIM + tid], acc);
            bbS[tid] = 1.7159f * fast_tanh(0.666f * acc);
        }
        __syncthreads();

        // ---- S4: heads = bb @ Wcat, split-K by 4 (chains of 32 FMAs) ----
        if (tid < 4 * HEADS) {
            const int c = tid / HEADS, o = tid % HEADS;
            const int k0 = c * 32;
            float acc = 0.0f;
            #pragma unroll
            for (int k = 0; k < 32; ++k)
                acc = fmaf(bbS[k0 + k], Wcat[(k0 + k) * HEADS + o], acc);
            part[c * HEADS + o] = acc;
        }
        __syncthreads();

        // ---- S5: reduce partials, gates, h update ----
        if (tid < UNITS) {
            float f1 = biasC[tid] +
                part[tid] + part[HEADS + tid] + part[2*HEADS + tid] + part[3*HEADS + tid];
            float f2 = biasC[50 + tid] +
                part[50 + tid] + part[HEADS + 50 + tid] + part[2*HEADS + 50 + tid] + part[3*HEADS + 50 + tid];
            float ta = biasC[100 + tid] +
                part[100 + tid] + part[HEADS + 100 + tid] + part[2*HEADS + 100 + tid] + part[3*HEADS + 100 + tid];
            float tb = biasC[150 + tid] +
                part[150 + tid] + part[HEADS + 150 + tid] + part[2*HEADS + 150 + tid] + part[3*HEADS + 150 + tid];
            const float ff1 = fast_tanh(f1);
            const float ff2 = fast_tanh(f2);
            const float s   = fast_sigmoid(ta + tb);
            hS[tid] = ff1 + s * (ff2 - ff1);   // ff1*(1-s) + s*ff2
        }
        __syncthreads();

        preReg = preNext;
    }

    // ---- Readout: out = h @ W_out + b_out (scalar) ----
    if (tid == 0) {
        float acc = bout[0];
        for (int i = 0; i < UNITS; ++i)
            acc = fmaf(hS[i], Wout[i], acc);
        out[0] = acc;
    }
}

// ---------------------------------------------------------------------------
// Host-side launch
// ---------------------------------------------------------------------------
extern "C" void kernel_launch(void* const* d_in, const int* in_sizes, int n_in,
                              void* d_out, int out_size, void* d_ws, size_t ws_size,
                              hipStream_t stream) {
    const float* x    = (const float*)d_in[0];   // [256, 2048, 64]
    const float* Wbb  = (const float*)d_in[1];   // [114, 128]
    const float* b_bb = (const float*)d_in[2];   // [128]
    const float* Wff1 = (const float*)d_in[3];
    const float* bff1 = (const float*)d_in[4];
    const float* Wff2 = (const float*)d_in[5];
    const float* bff2 = (const float*)d_in[6];
    const float* Wta  = (const float*)d_in[7];
    const float* bta  = (const float*)d_in[8];
    const float* Wtb  = (const float*)d_in[9];
    const float* btb  = (const float*)d_in[10];
    const float* Wout = (const float*)d_in[11];
    const float* bout = (const float*)d_in[12];
    float* out = (float*)d_out;

    // Only batch row 255 feeds the output.
    const float* x255 = x + (size_t)255 * T_STEPS * IN_DIM;

    float* Pre = (float*)d_ws;                   // 2048*128 floats = 1 MB scratch

    // Phase 1: 1024 tiles (128 M-tiles x 8 N-tiles), 8 waves per 256-thread block.
    cfc_pre_wmma<<<128, 256, 0, stream>>>(x255, Wbb, b_bb, Pre);

    // Phase 2: single workgroup, 129.7 KB dynamic LDS.
    const size_t lds_bytes =
        (UNITS * BB_DIM + BB_DIM * HEADS + HEADS + 64 + BB_DIM + 4 * HEADS) * sizeof(float);
    cfc_scan<<<1, SCAN_THREADS, lds_bytes, stream>>>(
        Pre, Wbb, Wff1, bff1, Wff2, bff2, Wta, bta, Wtb, btb, Wout, bout, out);
}